// MotionMamba_24816321036296
// MI455X (gfx1250) — compile-verified
//
#include <hip/hip_runtime.h>
#include <hip/hip_bf16.h>
#include <cstddef>
#include <cstdint>

// ---------------------------------------------------------------------------
// MotionMamba forward for MI455X (gfx1250, wave32, WMMA).
// GEMMs: V_WMMA_F32_16X16X4_F32 fed from LDS, tiles staged with
// GLOBAL_LOAD_ASYNC_TO_LDS_B128 (double-buffered, ASYNCcnt-tracked).
// Wave tile 32x64 (8 WMMA / k-step from 6 LDS loads); block tile 128x64.
// ---------------------------------------------------------------------------

typedef __attribute__((ext_vector_type(2))) float v2f;
typedef __attribute__((ext_vector_type(8))) float v8f;

#define ACT_NONE     0
#define ACT_SILU     1
#define ACT_GELU     2
#define ACT_SOFTPLUS 3

__device__ __forceinline__ float apply_act(float v, int act) {
    switch (act) {
        case ACT_SILU:     return v / (1.f + __expf(-v));
        case ACT_GELU:     return 0.5f * v * (1.f + erff(v * 0.70710678118654752f));
        case ACT_SOFTPLUS: return (v > 20.f) ? v : log1pf(__expf(v));
        default:           return v;
    }
}

// Async global -> LDS copy, 16 bytes per lane (ISA 10.x, ASYNCcnt).
__device__ __forceinline__ void async_copy_b128(uint32_t lds_off, const float* gptr) {
    asm volatile("global_load_async_to_lds_b128 %0, %1, off"
                 :: "v"(lds_off), "v"((unsigned long long)(uintptr_t)gptr)
                 : "memory");
}
__device__ __forceinline__ void wait_async_all() {
    asm volatile("s_wait_asynccnt 0x0" ::: "memory");
}

// ---------------------------------------------------------------------------
// GEMM: C[M,N] = act(A[M,K] @ W[N,K]^T + bias[N]) + Rsd[M,N]
// Block = 128 threads (4 waves). Block tile 128(M) x 64(N); wave tile 32 x 64.
// K consumed in KC=32 chunks, double-buffered through LDS via async copies.
// Fragment layout per CDNA5 ISA 7.12.2:
//   A 16x4 : lane(l): m = l&15 ; vgpr j holds K = j + 2*(l>>4)
//   B 4x16 : lane(l): n = l&15 ; vgpr j holds K = j + 2*(l>>4)
//   C 16x16: vgpr r : lanes 0-15 -> (M=r, N=lane); lanes 16-31 -> (M=r+8)
// ---------------------------------------------------------------------------
#define GEMM_BM 128
#define GEMM_BN 64
#define GEMM_KC 32
#define GEMM_LDST 36   // padded LDS row stride (floats): 16B-aligned, bank-safe
#define LDS_A_FLOATS (GEMM_BM * GEMM_LDST)
#define LDS_W_FLOATS (GEMM_BN * GEMM_LDST)
#define LDS_STAGE    (LDS_A_FLOATS + LDS_W_FLOATS)

__global__ __launch_bounds__(128) void gemm_f32_wmma_kernel(
    const float* __restrict__ A, int lda,
    const float* __restrict__ W, int ldw,
    const float* __restrict__ bias,
    const float* __restrict__ Rsd,
    float* __restrict__ C, int ldc,
    int M, int N, int K, int act)
{
    __shared__ float lds[2][LDS_STAGE];   // [stage][A tile | W tile]

    const int tid  = threadIdx.x;
    const int wave = tid >> 5;
    const int lane = tid & 31;
    const int lo   = lane & 15;
    const int hi   = lane >> 4;

    const int mBase = blockIdx.y * GEMM_BM;
    const int nBase = blockIdx.x * GEMM_BN;
    const int mWave = mBase + wave * 32;

    // Issue async copies for one K-chunk. A: 128 rows x 8 16B-chunks (8/thread);
    // W: 64 rows x 8 chunks (4/thread).  kc = valid K columns from k0.
    auto stage_in = [&](int stg, int k0, int kc) {
        #pragma unroll
        for (int q = 0; q < 8; ++q) {
            const int chunk = tid + 128 * q;   // 0..1023
            const int row   = chunk >> 3;
            const int g     = chunk & 7;
            if (g * 4 < kc) {
                int mr = mBase + row; if (mr >= M) mr = M - 1;
                uint32_t off = (uint32_t)(uintptr_t)&lds[stg][row * GEMM_LDST + g * 4];
                async_copy_b128(off, A + (size_t)mr * lda + k0 + g * 4);
            }
        }
        #pragma unroll
        for (int q = 0; q < 4; ++q) {
            const int chunk = tid + 128 * q;   // 0..511
            const int row   = chunk >> 3;
            const int g     = chunk & 7;
            if (g * 4 < kc) {
                int nr = nBase + row; if (nr >= N) nr = N - 1;
                uint32_t off = (uint32_t)(uintptr_t)&lds[stg][LDS_A_FLOATS + row * GEMM_LDST + g * 4];
                async_copy_b128(off, W + (size_t)nr * ldw + k0 + g * 4);
            }
        }
    };

    const int nChunks = (K + GEMM_KC - 1) / GEMM_KC;
    stage_in(0, 0, (K < GEMM_KC) ? K : GEMM_KC);

    const v8f vzero = {0.f,0.f,0.f,0.f,0.f,0.f,0.f,0.f};
    v8f acc[2][4] = {{vzero, vzero, vzero, vzero}, {vzero, vzero, vzero, vzero}};

    const int rowA0 = (wave * 32 +      lo) * GEMM_LDST;
    const int rowA1 = (wave * 32 + 16 + lo) * GEMM_LDST;

    for (int ci = 0; ci < nChunks; ++ci) {
        wait_async_all();        // this wave's async stores to LDS done
        __syncthreads();         // all waves' stores visible; prev compute done

        const int kNext = (ci + 1) * GEMM_KC;
        if (ci + 1 < nChunks) {
            int kcN = K - kNext; if (kcN > GEMM_KC) kcN = GEMM_KC;
            stage_in((ci + 1) & 1, kNext, kcN);   // prefetch next chunk
        }

        const int stg = ci & 1;
        int kc = K - ci * GEMM_KC; if (kc > GEMM_KC) kc = GEMM_KC;
        const float* __restrict__ At = &lds[stg][0];
        const float* __restrict__ Wt = &lds[stg][LDS_A_FLOATS];

        auto kstep = [&](int kk) {
            const int col = kk + 2 * hi;
            v2f a0; a0.x = At[rowA0 + col];  a0.y = At[rowA0 + col + 1];
            v2f a1; a1.x = At[rowA1 + col];  a1.y = At[rowA1 + col + 1];
            v2f b0; b0.x = Wt[(  0 + lo) * GEMM_LDST + col];  b0.y = Wt[(  0 + lo) * GEMM_LDST + col + 1];
            v2f b1; b1.x = Wt[( 16 + lo) * GEMM_LDST + col];  b1.y = Wt[( 16 + lo) * GEMM_LDST + col + 1];
            v2f b2; b2.x = Wt[( 32 + lo) * GEMM_LDST + col];  b2.y = Wt[( 32 + lo) * GEMM_LDST + col + 1];
            v2f b3; b3.x = Wt[( 48 + lo) * GEMM_LDST + col];  b3.y = Wt[( 48 + lo) * GEMM_LDST + col + 1];
            acc[0][0] = __builtin_amdgcn_wmma_f32_16x16x4_f32(false, a0, false, b0, (short)0, acc[0][0], false, false);
            acc[0][1] = __builtin_amdgcn_wmma_f32_16x16x4_f32(false, a0, false, b1, (short)0, acc[0][1], false, false);
            acc[0][2] = __builtin_amdgcn_wmma_f32_16x16x4_f32(false, a0, false, b2, (short)0, acc[0][2], false, false);
            acc[0][3] = __builtin_amdgcn_wmma_f32_16x16x4_f32(false, a0, false, b3, (short)0, acc[0][3], false, false);
            acc[1][0] = __builtin_amdgcn_wmma_f32_16x16x4_f32(false, a1, false, b0, (short)0, acc[1][0], false, false);
            acc[1][1] = __builtin_amdgcn_wmma_f32_16x16x4_f32(false, a1, false, b1, (short)0, acc[1][1], false, false);
            acc[1][2] = __builtin_amdgcn_wmma_f32_16x16x4_f32(false, a1, false, b2, (short)0, acc[1][2], false, false);
            acc[1][3] = __builtin_amdgcn_wmma_f32_16x16x4_f32(false, a1, false, b3, (short)0, acc[1][3], false, false);
        };

        if (kc == GEMM_KC) {
            // fast path: fully unrolled 8 k-steps so ds_loads pipeline ahead
            #pragma unroll
            for (int kk = 0; kk < GEMM_KC; kk += 4) kstep(kk);
        } else {
            for (int kk = 0; kk < kc; kk += 4) kstep(kk);
        }
    }

#pragma unroll
    for (int s = 0; s < 2; ++s) {
#pragma unroll
        for (int t = 0; t < 4; ++t) {
            const int col = nBase + 16 * t + lo;
            if (col >= N) continue;
            const float bv = bias ? bias[col] : 0.f;
#pragma unroll
            for (int r = 0; r < 8; ++r) {
                const int row = mWave + 16 * s + r + 8 * hi;
                if (row >= M) continue;
                float v = acc[s][t][r] + bv;
                v = apply_act(v, act);
                if (Rsd) v += Rsd[(size_t)row * ldc + col];
                C[(size_t)row * ldc + col] = v;
            }
        }
    }
}

// ---------------------------------------------------------------------------
// LayerNorm: one wave per row (d <= 512)
// ---------------------------------------------------------------------------
__global__ __launch_bounds__(256) void ln_kernel(
    const float* __restrict__ X, const float* __restrict__ g,
    const float* __restrict__ b, float* __restrict__ Y, int M, int d)
{
    const int row  = blockIdx.x * (blockDim.x >> 5) + (threadIdx.x >> 5);
    const int lane = threadIdx.x & 31;
    if (row >= M) return;
    const float* __restrict__ x = X + (size_t)row * d;

    float s = 0.f;
    for (int i = lane; i < d; i += 32) s += x[i];
#pragma unroll
    for (int off = 16; off > 0; off >>= 1) s += __shfl_xor(s, off, 32);
    const float mean = s / (float)d;

    float v = 0.f;
    for (int i = lane; i < d; i += 32) { float t = x[i] - mean; v += t * t; }
#pragma unroll
    for (int off = 16; off > 0; off >>= 1) v += __shfl_xor(v, off, 32);
    const float rstd = rsqrtf(v / (float)d + 1e-6f);

    for (int i = lane; i < d; i += 32)
        Y[(size_t)row * d + i] = (x[i] - mean) * rstd * g[i] + b[i];
}

// ---------------------------------------------------------------------------
// Causal depthwise conv (D_CONV=4) + SiLU.  u lives inside xz (stride ldxz).
// ---------------------------------------------------------------------------
__global__ __launch_bounds__(256) void conv_silu_kernel(
    const float* __restrict__ xz, int ldxz,
    const float* __restrict__ cw, const float* __restrict__ cb,
    float* __restrict__ uc, int Bt, int L, int dch)
{
    const long long idx = (long long)blockIdx.x * blockDim.x + threadIdx.x;
    const long long total = (long long)Bt * L * dch;
    if (idx >= total) return;
    const int ch = (int)(idx % dch);
    const int t  = (int)(idx / dch);
    const int l  = t % L;
    const int bq = t / L;

    float acc = cb[ch];
    const float* __restrict__ wr = cw + (size_t)ch * 4;
#pragma unroll
    for (int k = 0; k < 4; ++k) {
        const int lsrc = l + k - 3;
        if (lsrc >= 0)
            acc += xz[(size_t)(bq * L + lsrc) * ldxz + ch] * wr[k];
    }
    uc[(size_t)t * dch + ch] = acc / (1.f + __expf(-acc));
}

// ---------------------------------------------------------------------------
// Selective scan: one thread per (sequence, channel); 16 SSM states in regs.
// Fuses + u*D and * silu(z); writes y in place over uc (thread-exclusive col).
// ---------------------------------------------------------------------------
__global__ __launch_bounds__(256) void scan_kernel(
    const float* __restrict__ dt,   // (T, dch)
    const float* __restrict__ dbl, int ldd, int dtr,  // B at col dtr, C at dtr+16
    const float* __restrict__ uc,   // (T, dch)
    const float* __restrict__ xz, int ldxz, int zoff, // z = xz[:, zoff + ch]
    const float* __restrict__ A_log, const float* __restrict__ Dp,
    float* __restrict__ y,          // (T, dch)  (may alias uc)
    int Bt, int L, int dch)
{
    const long long idx = (long long)blockIdx.x * blockDim.x + threadIdx.x;
    if (idx >= (long long)Bt * dch) return;
    const int ch = (int)(idx % dch);
    const int bq = (int)(idx / dch);

    float a[16], h[16];
#pragma unroll
    for (int s = 0; s < 16; ++s) { a[s] = -__expf(A_log[(size_t)ch * 16 + s]); h[s] = 0.f; }
    const float Dv = Dp[ch];

    for (int l = 0; l < L; ++l) {
        const int t = bq * L + l;
        const float dtv = dt[(size_t)t * dch + ch];
        const float uv  = uc[(size_t)t * dch + ch];
        const float du  = dtv * uv;
        const float* __restrict__ Brow = dbl + (size_t)t * ldd + dtr;
        const float* __restrict__ Crow = Brow + 16;
        float accv = 0.f;
#pragma unroll
        for (int s = 0; s < 16; ++s) {
            h[s] = __expf(dtv * a[s]) * h[s] + du * Brow[s];
            accv += h[s] * Crow[s];
        }
        const float zv = xz[(size_t)t * ldxz + zoff + ch];
        y[(size_t)t * dch + ch] = (accv + uv * Dv) * (zv / (1.f + __expf(-zv)));
    }
}

// ---------------------------------------------------------------------------
// (b, F, N, c) -> (b, N, F, c)  (swap middle dims)
// ---------------------------------------------------------------------------
__global__ __launch_bounds__(256) void transpose_kernel(
    const float* __restrict__ in, float* __restrict__ out,
    int B, int F, int N, int C)
{
    const long long idx = (long long)blockIdx.x * blockDim.x + threadIdx.x;
    const long long total = (long long)B * F * N * C;
    if (idx >= total) return;
    const int c = (int)(idx % C);
    long long r = idx / C;
    const int j = (int)(r % N); r /= N;
    const int i = (int)(r % F);
    const int b = (int)(r / F);
    out[(((size_t)(b * N + j) * F + i) * C) + c] = in[idx];
}

// ---------------------------------------------------------------------------
// Embedding: out[t, o] = sum_{k<3} x[t,k] * W[o,k] + b[o]
// ---------------------------------------------------------------------------
__global__ __launch_bounds__(256) void emb_kernel(
    const float* __restrict__ x, const float* __restrict__ W,
    const float* __restrict__ b, float* __restrict__ out, int T, int N)
{
    const long long idx = (long long)blockIdx.x * blockDim.x + threadIdx.x;
    if (idx >= (long long)T * N) return;
    const int o = (int)(idx % N);
    const int t = (int)(idx / N);
    const float* xr = x + (size_t)t * 3;
    const float* wr = W + (size_t)o * 3;
    out[idx] = xr[0] * wr[0] + xr[1] * wr[1] + xr[2] * wr[2] + b[o];
}

// ---------------------------------------------------------------------------
// Host-side orchestration
// ---------------------------------------------------------------------------

// jax pytree flatten order (dict keys sorted, lists in order), 'params' then 'x':
enum {
    I_C6B = 0, I_C6W, I_C7B, I_C7W, I_EMB_B, I_EMB_W,
    I_PROJ_B0 = 6,   // proj_b[0], proj_b[1]
    I_PROJ_W0 = 8,   // proj_w[0], proj_w[1]
    I_SNB0 = 10,     // snb[0..2]
    I_SNG0 = 13,     // sng[0..2]
    I_STE0 = 16,     // ste[0..2], 17 leaves each
    I_TNB0 = 67,
    I_TNG0 = 70,
    I_TTE0 = 73,     // tte[0..2], 17 leaves each
    I_X    = 124
};
// per-block leaf offsets (sorted keys: fc1b,fc1w,fc2b,fc2w, mamba{A_log,D,conv_b,conv_w,dt_b,dt_w,in_proj,out_proj,x_proj}, n1b,n1g,n2b,n2g)
enum {
    B_FC1B = 0, B_FC1W, B_FC2B, B_FC2W,
    B_ALOG, B_D, B_CONVB, B_CONVW, B_DTB, B_DTW, B_INPROJ, B_OUTPROJ, B_XPROJ,
    B_N1B, B_N1G, B_N2B, B_N2G
};

static inline void launch_gemm(hipStream_t s,
    const float* A, int lda, const float* W, int ldw,
    const float* bias, const float* rsd,
    float* C, int ldc, int M, int N, int K, int act)
{
    dim3 grid((N + GEMM_BN - 1) / GEMM_BN, (M + GEMM_BM - 1) / GEMM_BM);
    gemm_f32_wmma_kernel<<<grid, dim3(128), 0, s>>>(A, lda, W, ldw, bias, rsd, C, ldc, M, N, K, act);
}

static inline void launch_ln(hipStream_t s, const float* X, const float* g,
                             const float* b, float* Y, int M, int d)
{
    const int waves = 8;
    ln_kernel<<<(M + waves - 1) / waves, waves * 32, 0, s>>>(X, g, b, Y, M, d);
}

static void run_block(hipStream_t s, float* X, int Bt, int L, int c,
                      void* const* d_in, int base,
                      float* bufH, float* bufXZ, float* bufUC,
                      float* bufDBL, float* bufDT)
{
    const int T    = Bt * L;
    const int dch  = 2 * c;             // din
    const int dtr  = (c + 15) / 16;
    const int ldd  = dtr + 32;

    const float* n1g   = (const float*)d_in[base + B_N1G];
    const float* n1b   = (const float*)d_in[base + B_N1B];
    const float* n2g   = (const float*)d_in[base + B_N2G];
    const float* n2b   = (const float*)d_in[base + B_N2B];
    const float* inprj = (const float*)d_in[base + B_INPROJ];
    const float* outpj = (const float*)d_in[base + B_OUTPROJ];
    const float* xprj  = (const float*)d_in[base + B_XPROJ];
    const float* dtw   = (const float*)d_in[base + B_DTW];
    const float* dtb   = (const float*)d_in[base + B_DTB];
    const float* convw = (const float*)d_in[base + B_CONVW];
    const float* convb = (const float*)d_in[base + B_CONVB];
    const float* alog  = (const float*)d_in[base + B_ALOG];
    const float* Dp    = (const float*)d_in[base + B_D];
    const float* fc1w  = (const float*)d_in[base + B_FC1W];
    const float* fc1b  = (const float*)d_in[base + B_FC1B];
    const float* fc2w  = (const float*)d_in[base + B_FC2W];
    const float* fc2b  = (const float*)d_in[base + B_FC2B];

    // 1. h = LN(x)
    launch_ln(s, X, n1g, n1b, bufH, T, c);
    // 2. xz = h @ in_proj^T  (N = 4c, no bias)
    launch_gemm(s, bufH, c, inprj, c, nullptr, nullptr, bufXZ, 4 * c, T, 4 * c, c, ACT_NONE);
    // 3. uc = silu(causal_conv(u) + conv_b)  (u = xz[:, :2c])
    {
        long long tot = (long long)T * dch;
        conv_silu_kernel<<<(unsigned)((tot + 255) / 256), 256, 0, s>>>(
            bufXZ, 4 * c, convw, convb, bufUC, Bt, L, dch);
    }
    // 4. dbl = uc @ x_proj^T  (N = dtr + 32)
    launch_gemm(s, bufUC, dch, xprj, dch, nullptr, nullptr, bufDBL, ldd, T, ldd, dch, ACT_NONE);
    // 5. dt = softplus(dbl[:, :dtr] @ dt_w^T + dt_b)
    launch_gemm(s, bufDBL, ldd, dtw, dtr, dtb, nullptr, bufDT, dch, T, dch, dtr, ACT_SOFTPLUS);
    // 6. selective scan; fused y = (ys + uc*D) * silu(z); y overwrites uc
    {
        long long tot = (long long)Bt * dch;
        scan_kernel<<<(unsigned)((tot + 255) / 256), 256, 0, s>>>(
            bufDT, bufDBL, ldd, dtr, bufUC, bufXZ, 4 * c, dch, alog, Dp,
            bufUC, Bt, L, dch);
    }
    // 7. x = x + y @ out_proj^T   (residual fused)
    launch_gemm(s, bufUC, dch, outpj, dch, nullptr, X, X, c, T, c, dch, ACT_NONE);
    // 8. h = LN(x)
    launch_ln(s, X, n2g, n2b, bufH, T, c);
    // 9. h2 = gelu(h @ fc1^T + fc1b)  -> reuse bufDT (T x 2c)
    launch_gemm(s, bufH, c, fc1w, c, fc1b, nullptr, bufDT, 2 * c, T, 2 * c, c, ACT_GELU);
    // 10. x = x + h2 @ fc2^T + fc2b   (residual fused)
    launch_gemm(s, bufDT, 2 * c, fc2w, 2 * c, fc2b, X, X, c, T, c, 2 * c, ACT_NONE);
}

extern "C" void kernel_launch(void* const* d_in, const int* in_sizes, int n_in,
                              void* d_out, int out_size, void* d_ws, size_t ws_size,
                              hipStream_t stream)
{
    if (n_in < 125) return;

    const int Bsz = 4, Fdim = 81, Ndim = 17;
    const int T = Bsz * Fdim * Ndim;   // 5508 tokens

    // workspace carve-up (floats)
    float* ws     = (float*)d_ws;
    float* bufX   = ws;                               // T*512
    float* bufT2  = bufX   + (size_t)T * 512;         // T*512
    float* bufH   = bufT2  + (size_t)T * 512;         // T*512
    float* bufXZ  = bufH   + (size_t)T * 512;         // T*2048
    float* bufUC  = bufXZ  + (size_t)T * 2048;        // T*1024
    float* bufDBL = bufUC  + (size_t)T * 1024;        // T*64
    float* bufDT  = bufDBL + (size_t)T * 64;          // T*1024

    const float* x_in  = (const float*)d_in[I_X];
    const float* embw  = (const float*)d_in[I_EMB_W];
    const float* embb  = (const float*)d_in[I_EMB_B];

    // embedding: (T,3) -> (T,512)
    {
        long long tot = (long long)T * 512;
        emb_kernel<<<(unsigned)((tot + 255) / 256), 256, 0, stream>>>(
            x_in, embw, embb, bufX, T, 512);
    }

    float* cur = bufX;
    float* alt = bufT2;
    int c = 512;

    for (int i = 0; i < 3; ++i) {
        const int steBase = I_STE0 + 17 * i;
        const int tteBase = I_TTE0 + 17 * i;

        // spatial block: (b*f, n, c)  Bt=324, L=17
        run_block(stream, cur, Bsz * Fdim, Ndim, c, d_in, steBase,
                  bufH, bufXZ, bufUC, bufDBL, bufDT);
        launch_ln(stream, cur, (const float*)d_in[I_SNG0 + i],
                  (const float*)d_in[I_SNB0 + i], cur, T, c);   // in-place (one wave per row)

        // (b,f,n,c) -> (b,n,f,c)
        {
            long long tot = (long long)T * c;
            transpose_kernel<<<(unsigned)((tot + 255) / 256), 256, 0, stream>>>(
                cur, alt, Bsz, Fdim, Ndim, c);
        }

        // temporal block: (b*n, f, c)  Bt=68, L=81
        run_block(stream, alt, Bsz * Ndim, Fdim, c, d_in, tteBase,
                  bufH, bufXZ, bufUC, bufDBL, bufDT);
        launch_ln(stream, alt, (const float*)d_in[I_TNG0 + i],
                  (const float*)d_in[I_TNB0 + i], alt, T, c);

        // (b,n,f,c) -> (b,f,n,c)
        {
            long long tot = (long long)T * c;
            transpose_kernel<<<(unsigned)((tot + 255) / 256), 256, 0, stream>>>(
                alt, cur, Bsz, Ndim, Fdim, c);
        }

        if (i < 2) {
            const float* pw = (const float*)d_in[I_PROJ_W0 + i];
            const float* pb = (const float*)d_in[I_PROJ_B0 + i];
            launch_gemm(stream, cur, c, pw, c, pb, nullptr, alt, c / 2, T, c / 2, c, ACT_NONE);
            float* t = cur; cur = alt; alt = t;
            c /= 2;
        }
    }

    // heads: out6 (T,6) ++ out7 (T,7), concatenated flat in d_out
    float* out = (float*)d_out;
    launch_gemm(stream, cur, c, (const float*)d_in[I_C6W], c,
                (const float*)d_in[I_C6B], nullptr, out, 6, T, 6, c, ACT_NONE);
    launch_gemm(stream, cur, c, (const float*)d_in[I_C7W], c,
                (const float*)d_in[I_C7B], nullptr, out + (size_t)T * 6, 7, T, 7, c, ACT_NONE);
}